// Myweightcrossentropyloss_28303834480670
// MI455X (gfx1250) — compile-verified
//
#include <hip/hip_runtime.h>

// Problem constants (from setup_inputs): input [8,19,512,1024] f32, target [8,1,512,1024] i32
#define NB     8
#define NC     19
#define NH     512
#define NW     1024
#define KK     15
#define HALF   (KK / 2)          // 7
#define TILE_W 128               // one thread per output column
#define TILE_H 18                // output rows per band (32-row band - 14 halo)
#define BAND   32
#define LCOLS  142               // TILE_W + 2*HALF
#define LSTRIDE 144              // padded LDS row stride (words)

typedef __attribute__((ext_vector_type(2))) float v2f;
typedef __attribute__((ext_vector_type(8))) float v8f;

#if defined(__gfx1250__) && __has_builtin(__builtin_amdgcn_global_load_async_to_lds_b32)
#define HAVE_ASYNC_LDS 1
#endif
#if defined(__gfx1250__) && __has_builtin(__builtin_amdgcn_wmma_f32_16x16x4_f32)
#define HAVE_WMMA4 1
#endif

__device__ __forceinline__ void wait_async_all() {
#if defined(__gfx1250__)
#if __has_builtin(__builtin_amdgcn_s_wait_asynccnt)
  __builtin_amdgcn_s_wait_asynccnt(0);
#else
  asm volatile("s_wait_asynccnt 0x0" ::: "memory");
#endif
#endif
}

__global__ __launch_bounds__(TILE_W) void loss_main(
    const float* __restrict__ input, const int* __restrict__ target,
    float* __restrict__ partial)
{
  __shared__ unsigned tile[BAND][LSTRIDE];   // raw class band (int32 classes)
  __shared__ unsigned bsl[5][LSTRIDE];       // vertical bit-slices of class id
  __shared__ float    lut[KK * KK + 1];      // lut[n] = -225/n (exact f32 divide)
  __shared__ float    red[TILE_W];
  __shared__ float    red16[16];

  const int tid = threadIdx.x;
  const int w0  = blockIdx.x * TILE_W;
  const int h0  = blockIdx.y * TILE_H;
  const int bz  = blockIdx.z;

  for (int i = tid; i <= KK * KK; i += TILE_W)
    lut[i] = i ? (-(float)(KK * KK) / (float)i) : 0.0f;

  // ---- Phase 1: stage 32 x 142 class band into LDS (async DMA on gfx1250) ----
  const int rbase = h0 - HALF;
  const int cbase = w0 - HALF;
  const int* tb = target + (size_t)bz * NH * NW;
  for (int idx = tid; idx < BAND * LCOLS; idx += TILE_W) {
    const int r = idx / LCOLS, c = idx - r * LCOLS;
    const int gr = rbase + r, gc = cbase + c;
    if (gr < 0 || gr >= NH || gc < 0 || gc >= NW) {
      tile[r][c] = 31u;   // sentinel class: never equals a real class (0..18) -> zero-pad semantics
    } else {
#if defined(HAVE_ASYNC_LDS)
      __builtin_amdgcn_global_load_async_to_lds_b32(
          (__attribute__((address_space(1))) int*)(tb + (size_t)gr * NW + gc),
          (__attribute__((address_space(3))) int*)&tile[r][c], 0, 0);
#else
      tile[r][c] = (unsigned)tb[(size_t)gr * NW + gc];
#endif
    }
  }
  wait_async_all();
  __syncthreads();

  // ---- Phase 1b: vertical bit-slice: bsl[i][c] bit r = bit i of class(r,c) ----
  for (int c = tid; c < LCOLS; c += TILE_W) {
    unsigned b0 = 0, b1 = 0, b2 = 0, b3 = 0, b4 = 0;
#pragma unroll
    for (int r = 0; r < BAND; ++r) {
      const unsigned v = tile[r][c];
      b0 |= (v        & 1u) << r;
      b1 |= ((v >> 1) & 1u) << r;
      b2 |= ((v >> 2) & 1u) << r;
      b3 |= ((v >> 3) & 1u) << r;
      b4 |= ((v >> 4) & 1u) << r;
    }
    bsl[0][c] = b0; bsl[1][c] = b1; bsl[2][c] = b2; bsl[3][c] = b3; bsl[4][c] = b4;
  }
  __syncthreads();

  // ---- Phase 2: per-pixel same-class window count via bitplane popcount ----
  unsigned W0[KK], W1[KK], W2[KK], W3[KK], W4[KK];
#pragma unroll
  for (int dx = 0; dx < KK; ++dx) {
    W0[dx] = bsl[0][tid + dx];
    W1[dx] = bsl[1][tid + dx];
    W2[dx] = bsl[2][tid + dx];
    W3[dx] = bsl[3][tid + dx];
    W4[dx] = bsl[4][tid + dx];
  }

  float acc = 0.0f;
  const float* ib = input + (size_t)bz * NC * NH * NW;
  const int wcol = w0 + tid;
  for (int row = 0; row < TILE_H; ++row) {
    const int h = h0 + row;
    if (h >= NH) break;                       // uniform across the block
    const int bi = row + HALF;                // band index of center row
    const unsigned t =
        ((W0[HALF] >> bi) & 1u)        | (((W1[HALF] >> bi) & 1u) << 1) |
        (((W2[HALF] >> bi) & 1u) << 2) | (((W3[HALF] >> bi) & 1u) << 3) |
        (((W4[HALF] >> bi) & 1u) << 4);
    const unsigned s0 = 0u - ( t       & 1u);
    const unsigned s1 = 0u - ((t >> 1) & 1u);
    const unsigned s2 = 0u - ((t >> 2) & 1u);
    const unsigned s3 = 0u - ((t >> 3) & 1u);
    const unsigned s4 = 0u - ((t >> 4) & 1u);
    const unsigned wmask = 0x7FFFu << row;    // rows h-7..h+7 within the band
    unsigned wsum = 0;
#pragma unroll
    for (int dx = 0; dx < KK; ++dx) {
      const unsigned m = ~((W0[dx] ^ s0) | (W1[dx] ^ s1) | (W2[dx] ^ s2) |
                           (W3[dx] ^ s3) | (W4[dx] ^ s4));
      wsum += __popc(m & wmask);
    }
    // gather score at the true class; streaming (no reuse, input > L2)
    const float v = __builtin_nontemporal_load(ib + ((size_t)t * NH + h) * NW + wcol);
    acc += v * lut[wsum];
  }

  // ---- Block reduction: 128 partials -> 1, via V_WMMA_F32_16X16X4_F32 ----
  red[tid] = acc;
  __syncthreads();
#if defined(HAVE_WMMA4)
  if (tid < 32) {                             // full wave: EXEC all ones for WMMA
    v2f ones; ones[0] = 1.0f; ones[1] = 1.0f; // A = all-ones 16x4 -> D[m][n] = sum_k B[k][n]
    v2f p0;   p0[0] = red[tid];      p0[1] = red[tid + 32];
    v2f p1;   p1[0] = red[tid + 64]; p1[1] = red[tid + 96];
    v8f c = {0.f, 0.f, 0.f, 0.f, 0.f, 0.f, 0.f, 0.f};
    c = __builtin_amdgcn_wmma_f32_16x16x4_f32(false, ones, false, p0, (short)0, c, false, false);
    c = __builtin_amdgcn_wmma_f32_16x16x4_f32(false, ones, false, p1, (short)0, c, false, false);
    if (tid < 16) red16[tid] = c[0];          // row M=0: 16 column sums
  }
  __syncthreads();
  if (tid == 0) {
    float s = 0.f;
#pragma unroll
    for (int i = 0; i < 16; ++i) s += red16[i];
    partial[((size_t)bz * gridDim.y + blockIdx.y) * gridDim.x + blockIdx.x] = s;
  }
#else
  if (tid == 0) {
    float s = 0.f;
    for (int i = 0; i < TILE_W; ++i) s += red[i];
    partial[((size_t)bz * gridDim.y + blockIdx.y) * gridDim.x + blockIdx.x] = s;
  }
#endif
}

__global__ __launch_bounds__(256) void loss_finalize(
    const float* __restrict__ partial, int n, float* __restrict__ out)
{
  __shared__ float s[256];
  float a = 0.f;
  for (int i = threadIdx.x; i < n; i += 256) a += partial[i];
  s[threadIdx.x] = a;
  __syncthreads();
  for (int off = 128; off > 0; off >>= 1) {
    if ((int)threadIdx.x < off) s[threadIdx.x] += s[threadIdx.x + off];
    __syncthreads();
  }
  if (threadIdx.x == 0)
    out[0] = s[0] * (1.0f / (float)((size_t)NB * NH * NW));
}

extern "C" void kernel_launch(void* const* d_in, const int* in_sizes, int n_in,
                              void* d_out, int out_size, void* d_ws, size_t ws_size,
                              hipStream_t stream) {
  (void)in_sizes; (void)n_in; (void)out_size; (void)ws_size;
  const float* input  = (const float*)d_in[0];
  const int*   target = (const int*)d_in[1];
  float* partial = (float*)d_ws;

  dim3 grid(NW / TILE_W, (NH + TILE_H - 1) / TILE_H, NB);   // 8 x 29 x 8 = 1856 blocks
  loss_main<<<grid, TILE_W, 0, stream>>>(input, target, partial);
  const int nblocks = (int)(grid.x * grid.y * grid.z);
  loss_finalize<<<1, 256, 0, stream>>>(partial, nblocks, (float*)d_out);
}